// DecoderRNN_10582799417875
// MI455X (gfx1250) — compile-verified
//
#include <hip/hip_runtime.h>
#include <math.h>

// Problem constants (reference: V, E, D, B, T = 32000, 256, 25, 64, 24)
#define BB   64
#define TT   24
#define EE   256
#define DD   25
#define HH   625          // D*D
#define INN  906          // E + H + D
#define VV   32000
#define G4H  2500         // 4*H
#define G4D  100          // 4*D
#define NBT  4            // N-tiles (16 cols each) per wave in blocked GEMM

typedef float v2f __attribute__((ext_vector_type(2)));
typedef float v8f __attribute__((ext_vector_type(8)));

__device__ __forceinline__ float sigf(float x) { return 1.0f / (1.0f + __expf(-x)); }

// ---------------------------------------------------------------------------
// zero-fill (graph-capture-safe replacement for memset)
// ---------------------------------------------------------------------------
__global__ void zero_f32(float* __restrict__ p, int n) {
    int i = blockIdx.x * blockDim.x + threadIdx.x;
    if (i < n) p[i] = 0.0f;
}

// ---------------------------------------------------------------------------
// One GEMM pass: c[j] += A * W[j]^T over K, software-pipelined.
// Ap / Wp[j] already include the per-lane half*2 element offset.
// Double-buffered staging registers (a_cur/b_cur vs a_nxt/b_nxt) let the
// compiler keep the next k-step's 5 loads in flight while the current
// 4 WMMAs execute (partial s_wait_loadcnt instead of full drains).
// ---------------------------------------------------------------------------
__device__ __forceinline__ void gemm_pass(v8f* c, const float* Ap,
                                          const float* const* Wp,
                                          int K, int half)
{
    const int kfull = K & ~3;
    if (kfull > 0) {
        v2f a_cur, b_cur[NBT];
        a_cur.x = Ap[0]; a_cur.y = Ap[1];
#pragma unroll
        for (int j = 0; j < NBT; ++j) {
            b_cur[j].x = Wp[j][0]; b_cur[j].y = Wp[j][1];
        }
        int k0 = 0;
        for (; k0 + 4 < kfull; k0 += 4) {
            // issue next k-step's loads first (distinct regs -> stay in flight)
            v2f a_nxt, b_nxt[NBT];
            a_nxt.x = Ap[k0 + 4]; a_nxt.y = Ap[k0 + 5];
#pragma unroll
            for (int j = 0; j < NBT; ++j) {
                b_nxt[j].x = Wp[j][k0 + 4]; b_nxt[j].y = Wp[j][k0 + 5];
            }
            // compute on current registers (loaded last iteration)
#pragma unroll
            for (int j = 0; j < NBT; ++j) {
                c[j] = __builtin_amdgcn_wmma_f32_16x16x4_f32(
                           false, a_cur, false, b_cur[j], (short)0, c[j],
                           false, false);
            }
            a_cur = a_nxt;
#pragma unroll
            for (int j = 0; j < NBT; ++j) b_cur[j] = b_nxt[j];
        }
        // epilogue: last full k-step
#pragma unroll
        for (int j = 0; j < NBT; ++j) {
            c[j] = __builtin_amdgcn_wmma_f32_16x16x4_f32(
                       false, a_cur, false, b_cur[j], (short)0, c[j],
                       false, false);
        }
    }
    if (kfull < K) {                       // guarded K tail (K % 4 != 0)
        const int k0 = kfull, ka = k0 + half * 2;
        v2f a;
        a.x = (ka     < K) ? Ap[k0]     : 0.0f;
        a.y = (ka + 1 < K) ? Ap[k0 + 1] : 0.0f;
#pragma unroll
        for (int j = 0; j < NBT; ++j) {
            v2f b;
            b.x = (ka     < K) ? Wp[j][k0]     : 0.0f;
            b.y = (ka + 1 < K) ? Wp[j][k0 + 1] : 0.0f;
            c[j] = __builtin_amdgcn_wmma_f32_16x16x4_f32(
                       false, a, false, b, (short)0, c[j], false, false);
        }
    }
}

// ---------------------------------------------------------------------------
// Register-blocked fp32 WMMA GEMM (one wave = 16 x 64 output tile):
//   Dst[M,N] = A1[M,K1] * W1[N,K1]^T  (+ A2[M,K2] * W2[N,K2]^T if K2>0)
//              + biasA[n] + biasB[n]
//
// V_WMMA_F32_16X16X4_F32 register layouts per CDNA5 ISA (CDNA5 ISA):
//   A 16x4 : lanes 0-15 -> rows m0+l, regs {K0,K1}; lanes 16-31 -> {K2,K3}
//   B 4x16 : lanes 0-15 -> col n, regs {K0,K1};     lanes 16-31 -> {K2,K3}
//   C/D    : reg v, lanes 0-15 -> (m0+v, n); lanes 16-31 -> (m0+v+8, n)
// A registers reused across NBT=4 WMMAs. EXEC stays all-ones
// (selects + uniform loop branches only), as WMMA requires.
// ---------------------------------------------------------------------------
__global__ __launch_bounds__(32)
void gemm_wmma_blk(const float* __restrict__ A1, int lda1,
                   const float* __restrict__ W1, int ldw1, int K1,
                   const float* __restrict__ A2, int lda2,
                   const float* __restrict__ W2, int ldw2, int K2,
                   int M, int N,
                   const float* __restrict__ biasA,
                   const float* __restrict__ biasB,
                   float* __restrict__ Dst, int ldd)
{
    const int lane = threadIdx.x;        // 0..31
    const int half = lane >> 4;          // 0 or 1
    const int l15  = lane & 15;
    const int m0   = blockIdx.y * 16;
    const int n0   = blockIdx.x * (16 * NBT);

    int n[NBT];
    const float* Wp1[NBT];
    const float* Wp2[NBT];
    v8f c[NBT];

#pragma unroll
    for (int j = 0; j < NBT; ++j) {
        n[j] = n0 + j * 16 + l15;
        float bv = 0.0f;
        if (n[j] < N) {
            if (biasA) bv += biasA[n[j]];
            if (biasB) bv += biasB[n[j]];
        }
#pragma unroll
        for (int v = 0; v < 8; ++v) c[j][v] = bv;
        const int ncl = min(n[j], N - 1);   // clamp: garbage cols are discarded
        Wp1[j] = W1 + (size_t)ncl * ldw1 + half * 2;
        Wp2[j] = (K2 > 0) ? (W2 + (size_t)ncl * ldw2 + half * 2) : Wp1[j];
    }
    const int arow = min(m0 + l15, M - 1);  // clamp: garbage rows are discarded

    // pass 1: A1 * W1^T
    gemm_pass(c, A1 + (size_t)arow * lda1 + half * 2, Wp1, K1, half);
    // pass 2 (optional): A2 * W2^T
    if (K2 > 0) {
        gemm_pass(c, A2 + (size_t)arow * lda2 + half * 2, Wp2, K2, half);
    }

    // ---- store ----
#pragma unroll
    for (int j = 0; j < NBT; ++j) {
#pragma unroll
        for (int v = 0; v < 8; ++v) {
            const int m = m0 + v + half * 8;
            if (m < M && n[j] < N) Dst[(size_t)m * ldd + n[j]] = c[j][v];
        }
    }
}

// ---------------------------------------------------------------------------
// Pack x_t = [cU(25) | word(256) | cS(625) | pad(2)] into X[64][908]
// t==0 uses embed_table[0] (word0); t>=1 uses captions[:, t]
// ---------------------------------------------------------------------------
__global__ void pack_x(float* __restrict__ X,
                       const float* __restrict__ cU,
                       const float* __restrict__ cS,
                       const float* __restrict__ embed,
                       const int*   __restrict__ captions,
                       int t)
{
    int i = blockIdx.x * blockDim.x + threadIdx.x;
    if (i >= BB * 908) return;
    int b = i / 908, k = i % 908;
    float v;
    if (k < DD) {
        v = cU[b * DD + k];
    } else if (k < DD + EE) {
        int w = (t == 0) ? 0 : captions[b * TT + t];
        v = embed[(size_t)w * EE + (k - DD)];
    } else if (k < INN) {
        v = cS[b * HH + (k - DD - EE)];
    } else {
        v = 0.0f;
    }
    X[i] = v;
}

// ---------------------------------------------------------------------------
// LSTM pointwise update for both cells (gate order i,f,g,o)
// ---------------------------------------------------------------------------
__global__ void lstm_update(const float* __restrict__ gS,
                            const float* __restrict__ gU,
                            float* __restrict__ hS, float* __restrict__ cS,
                            float* __restrict__ hU, float* __restrict__ cU)
{
    int i = blockIdx.x * blockDim.x + threadIdx.x;
    if (i >= BB * (HH + DD)) return;
    int b = i / (HH + DD), j = i % (HH + DD);
    if (j < HH) {
        const float* g = gS + (size_t)b * G4H;
        float gi = g[j], gf = g[HH + j], gg = g[2 * HH + j], go = g[3 * HH + j];
        float cn = sigf(gf) * cS[b * HH + j] + sigf(gi) * tanhf(gg);
        cS[b * HH + j] = cn;
        hS[b * HH + j] = sigf(go) * tanhf(cn);
    } else {
        int j2 = j - HH;
        const float* g = gU + (size_t)b * G4D;
        float gi = g[j2], gf = g[DD + j2], gg = g[2 * DD + j2], go = g[3 * DD + j2];
        float cn = sigf(gf) * cU[b * DD + j2] + sigf(gi) * tanhf(gg);
        cU[b * DD + j2] = cn;
        hU[b * DD + j2] = sigf(go) * tanhf(cn);
    }
}

// ut[b][j] = wub[j] + sum_k hU[b][k] * wuW[j][k]   (K = 25, tiny)
__global__ void ut_kernel(float* __restrict__ ut,
                          const float* __restrict__ hU,
                          const float* __restrict__ wuW,
                          const float* __restrict__ wub)
{
    int i = blockIdx.x * blockDim.x + threadIdx.x;
    if (i >= BB * HH) return;
    int b = i / HH, j = i % HH;
    float s = wub[j];
    const float* h = hU + b * DD;
    const float* w = wuW + (size_t)j * DD;
#pragma unroll
    for (int k = 0; k < DD; ++k) s += h[k] * w[k];
    ut[i] = s;
}

// ft[b][r][c] = sum_k ut[b][r][k] * hS[b][k][c]   (batched 25x25x25)
__global__ void ft_kernel(float* __restrict__ ftout,
                          const float* __restrict__ ut,
                          const float* __restrict__ hS)
{
    int i = blockIdx.x * blockDim.x + threadIdx.x;
    if (i >= BB * HH) return;
    int b = i / HH, rc = i % HH;
    int r = rc / DD, cc = rc % DD;
    const float* u = ut + (size_t)b * HH + r * DD;
    const float* m = hS + (size_t)b * HH + cc;
    float s = 0.0f;
#pragma unroll
    for (int k = 0; k < DD; ++k) s += u[k] * m[k * DD];
    ftout[(size_t)b * HH + rc] = s;
}

// ---------------------------------------------------------------------------
extern "C" void kernel_launch(void* const* d_in, const int* in_sizes, int n_in,
                              void* d_out, int out_size, void* d_ws, size_t ws_size,
                              hipStream_t stream)
{
    (void)in_sizes; (void)n_in; (void)out_size; (void)ws_size;

    const float* features = (const float*)d_in[0];
    const int*   captions = (const int*)  d_in[1];
    const float* embed    = (const float*)d_in[2];
    const float* WihS     = (const float*)d_in[3];
    const float* WhhS     = (const float*)d_in[4];
    const float* bihS     = (const float*)d_in[5];
    const float* bhhS     = (const float*)d_in[6];
    const float* WihU     = (const float*)d_in[7];
    const float* WhhU     = (const float*)d_in[8];
    const float* bihU     = (const float*)d_in[9];
    const float* bhhU     = (const float*)d_in[10];
    const float* fcW      = (const float*)d_in[11];
    const float* fcb      = (const float*)d_in[12];
    const float* szW      = (const float*)d_in[13];
    const float* szb      = (const float*)d_in[14];
    const float* wuW      = (const float*)d_in[15];
    const float* wub      = (const float*)d_in[16];
    float* out = (float*)d_out;

    // workspace layout (floats)
    float* ws = (float*)d_ws;
    float* X  = ws;                      // 64*908
    float* gS = X  + BB * 908;           // 64*2500
    float* gU = gS + BB * G4H;           // 64*100
    float* hS = gU + BB * G4D;           // 64*625
    float* cS = hS + BB * HH;            // 64*625
    float* hU = cS + BB * HH;            // 64*25
    float* cU = hU + BB * DD;            // 64*25
    float* ut = cU + BB * DD;            // 64*625
    float* WF = ut + BB * HH;            // 24*64*625 word-feature buffer

    const int thr = 256;
    // fresh state each call (ws not re-poisoned between replays)
    zero_f32<<<(BB * HH + thr - 1) / thr, thr, 0, stream>>>(hS, BB * HH);
    zero_f32<<<(BB * DD + thr - 1) / thr, thr, 0, stream>>>(hU, BB * DD);
    zero_f32<<<(BB * DD + thr - 1) / thr, thr, 0, stream>>>(cU, BB * DD);
    zero_f32<<<(BB * HH + thr - 1) / thr, thr, 0, stream>>>(WF, BB * HH); // wordOut[0]=0

    const int NW = 16 * NBT;             // 64 output cols per wave

    // cS0 = sStart = features @ szW^T + szb   (M=64, N=625, K=256)
    gemm_wmma_blk<<<dim3((HH + NW - 1) / NW, BB / 16), 32, 0, stream>>>(
        features, EE, szW, EE, EE,
        nullptr, 0, nullptr, 0, 0,
        BB, HH, szb, nullptr, cS, HH);

    const dim3 gridS((G4H + NW - 1) / NW, BB / 16);  // 40 x 4
    const dim3 gridU((G4D + NW - 1) / NW, BB / 16);  //  2 x 4

    for (int t = 0; t < TT - 1; ++t) {               // 23 sequential steps
        pack_x<<<(BB * 908 + thr - 1) / thr, thr, 0, stream>>>(
            X, cU, cS, embed, captions, t);

        // gS = X @ WihS^T + hS @ WhhS^T + (bihS + bhhS)   (fused two-pass)
        gemm_wmma_blk<<<gridS, 32, 0, stream>>>(
            X, 908, WihS, INN, INN,
            hS, HH, WhhS, HH, HH,
            BB, G4H, bihS, bhhS, gS, G4H);

        // gU = X @ WihU^T + hU @ WhhU^T + (bihU + bhhU)
        gemm_wmma_blk<<<gridU, 32, 0, stream>>>(
            X, 908, WihU, INN, INN,
            hU, DD, WhhU, DD, DD,
            BB, G4D, bihU, bhhU, gU, G4D);

        lstm_update<<<(BB * (HH + DD) + thr - 1) / thr, thr, 0, stream>>>(
            gS, gU, hS, cS, hU, cU);

        ut_kernel<<<(BB * HH + thr - 1) / thr, thr, 0, stream>>>(ut, hU, wuW, wub);
        ft_kernel<<<(BB * HH + thr - 1) / thr, thr, 0, stream>>>(
            WF + (size_t)(t + 1) * BB * HH, ut, hS);
    }

    // out[T*B, V] = WF[T*B, H] @ fcW^T + fcb   (M=1536, N=32000, K=625)
    gemm_wmma_blk<<<dim3(VV / NW, (TT * BB) / 16), 32, 0, stream>>>(
        WF, HH, fcW, HH, HH,
        nullptr, 0, nullptr, 0, 0,
        TT * BB, VV, fcb, nullptr, out, VV);
}